// AttnProcessor_LoRA_Capture_9775345566323
// MI455X (gfx1250) — compile-verified
//
#include <hip/hip_runtime.h>
#include <hip/hip_bf16.h>
#include <math.h>

typedef float v2f __attribute__((ext_vector_type(2)));
typedef float v8f __attribute__((ext_vector_type(8)));

#define B_ 16
#define H_ 10
#define L_ 4096
#define S_ 77
#define D_ 64
#define NE_ 20
#define SHRINK_ 2.0f
#define SCALE_ 0.125f   /* 1/sqrt(64) */

#define LTILE 128
#define NWAVE 8

#define QSTR 66   /* row-indexed-by-lane reads: stride%64==2 -> conflict-free */
#define KSTR 66
#define VSTR 72   /* col-indexed-by-lane reads, rows 2 apart -> +16 banks */
#define WSTR 84

static __device__ __forceinline__ v8f wmma_f32(v2f a, v2f b, v8f c) {
  return __builtin_amdgcn_wmma_f32_16x16x4_f32(false, a, false, b, (short)0, c, false, false);
}

/* ---------------- Kernel 1: subj(b,l,n) = mean_h score[b,h,l,idx_n[n]] ------- */
__global__ __launch_bounds__(256)
void subj_scores_kernel(const float* __restrict__ q, const float* __restrict__ kk_,
                        const int* __restrict__ idx_b, const int* __restrict__ idx_n,
                        float* __restrict__ subj)
{
  __shared__ float Qs[LTILE * QSTR];
  __shared__ float Ks[32 * KSTR];
  const int b   = blockIdx.y;
  const int lt  = blockIdx.x;
  const int tid = threadIdx.x;
  const int lane = tid & 31, wave = tid >> 5;
  const int lm = lane & 15, half = lane >> 4, klo = half * 2;
  const int l0 = lt * LTILE;
  const int bb0 = idx_b[b * NE_];

  v8f z = {0,0,0,0,0,0,0,0};
  v8f c0 = z, c1 = z;

  for (int h = 0; h < H_; ++h) {
    __syncthreads();
    const float* qbase = q + (((size_t)bb0 * H_ + h) * L_ + l0) * D_;
    for (int i = 0; i < 8; ++i) {
      int fi = tid + i * 256;
      int row = fi >> 4, c4 = (fi & 15) * 4;
      float4 t = *(const float4*)(qbase + row * D_ + c4);
      float* d = &Qs[row * QSTR + c4];
      d[0] = t.x; d[1] = t.y; d[2] = t.z; d[3] = t.w;
    }
    for (int i = 0; i < 2; ++i) {
      int fi = tid + i * 256;
      int row = fi >> 4, c4 = (fi & 15) * 4;
      float4 t = make_float4(0.f, 0.f, 0.f, 0.f);
      if (row < NE_) {
        int bb = idx_b[b * NE_ + row];
        int nr = idx_n[b * NE_ + row];
        t = *(const float4*)(kk_ + (((size_t)bb * H_ + h) * S_ + nr) * D_ + c4);
      }
      float* d = &Ks[row * KSTR + c4];
      d[0] = t.x; d[1] = t.y; d[2] = t.z; d[3] = t.w;
    }
    __syncthreads();
    const float* qw = &Qs[wave * 16 * QSTR];
    #pragma unroll
    for (int kstep = 0; kstep < D_; kstep += 4) {
      v2f a  = { qw[lm * QSTR + kstep + klo],        qw[lm * QSTR + kstep + klo + 1] };
      v2f b0 = { Ks[lm * KSTR + kstep + klo],        Ks[lm * KSTR + kstep + klo + 1] };
      v2f b1 = { Ks[(16 + lm) * KSTR + kstep + klo], Ks[(16 + lm) * KSTR + kstep + klo + 1] };
      c0 = wmma_f32(a, b0, c0);
      c1 = wmma_f32(a, b1, c1);
    }
  }
  const float f = SCALE_ / (float)H_;
  #pragma unroll
  for (int vv = 0; vv < 8; ++vv) {
    int rM = vv + half * 8;
    size_t l = (size_t)l0 + wave * 16 + rM;
    int n0 = lm;
    if (n0 < NE_) subj[((size_t)b * L_ + l) * NE_ + n0] = c0[vv] * f;
    int n1 = 16 + lm;
    if (n1 < NE_) subj[((size_t)b * L_ + l) * NE_ + n1] = c1[vv] * f;
  }
}

/* ---------------- Kernel 2: softmax over L + spatial moments ----------------- */
static __device__ __forceinline__ float blk_sum(float v, float* red, int tid) {
  red[tid] = v; __syncthreads();
  for (int s = 128; s > 0; s >>= 1) { if (tid < s) red[tid] += red[tid + s]; __syncthreads(); }
  float r = red[0]; __syncthreads();
  return r;
}

__global__ __launch_bounds__(256)
void moments_kernel(const float* __restrict__ subj, float* __restrict__ params)
{
  __shared__ float red[256];
  const int j = blockIdx.x, b = blockIdx.y;
  const int tid = threadIdx.x;
  const float* p = subj + (size_t)b * L_ * NE_ + j;

  float vals[16];
  float m = -INFINITY;
  #pragma unroll
  for (int i = 0; i < 16; ++i) {
    vals[i] = p[(size_t)(tid + i * 256) * NE_];
    m = fmaxf(m, vals[i]);
  }
  red[tid] = m; __syncthreads();
  for (int s = 128; s > 0; s >>= 1) { if (tid < s) red[tid] = fmaxf(red[tid], red[tid + s]); __syncthreads(); }
  m = red[0]; __syncthreads();

  float E = 0.f, Sx = 0.f, Sy = 0.f, Sxx = 0.f, Syy = 0.f;
  #pragma unroll
  for (int i = 0; i < 16; ++i) {
    int l = tid + i * 256;
    float e = expf(vals[i] - m);
    float x = (float)(l & 63), y = (float)(l >> 6);
    E += e; Sx += e * x; Sy += e * y; Sxx += e * x * x; Syy += e * y * y;
  }
  E   = blk_sum(E,   red, tid);
  Sx  = blk_sum(Sx,  red, tid);
  Sy  = blk_sum(Sy,  red, tid);
  Sxx = blk_sum(Sxx, red, tid);
  Syy = blk_sum(Syy, red, tid);

  if (tid == 0) {
    float xc = Sx / E, yc = Sy / E;
    float vx = Sxx / E - xc * xc;
    float vy = Syy / E - yc * yc;
    float sx = sqrtf(vx) / (SHRINK_ - 1.f);
    float sy = sqrtf(vy) / (SHRINK_ - 1.f);
    float i2x = 1.f / (2.f * sx * sx);
    float i2y = 1.f / (2.f * sy * sy);
    float dx = rintf(xc) - xc, dy = rintf(yc) - yc;
    float peak = expf(-(dx * dx * i2x + dy * dy * i2y));
    float* o = params + ((size_t)b * NE_ + j) * 5;
    o[0] = xc; o[1] = yc; o[2] = i2x; o[3] = i2y; o[4] = peak;
  }
}

/* ---------------- Kernel 3: global grid max ---------------------------------- */
__global__ __launch_bounds__(512)
void gmax_kernel(const float* __restrict__ params, float* __restrict__ g)
{
  __shared__ float red[512];
  int tid = threadIdx.x;
  float m = -INFINITY;
  for (int i = tid; i < B_ * NE_; i += 512) m = fmaxf(m, params[(size_t)i * 5 + 4]);
  red[tid] = m; __syncthreads();
  for (int s = 256; s > 0; s >>= 1) { if (tid < s) red[tid] = fmaxf(red[tid], red[tid + s]); __syncthreads(); }
  if (tid == 0) g[0] = red[0];
}

/* ---------------- Kernel 4: attention with gaussian scaling ------------------ */
__global__ __launch_bounds__(256)
void attn_kernel(const float* __restrict__ q, const float* __restrict__ kk_,
                 const float* __restrict__ vv_,
                 const int* __restrict__ idx_b, const int* __restrict__ idx_n,
                 const float* __restrict__ params, const float* __restrict__ gmax,
                 float* __restrict__ out)
{
  __shared__ float QW[NWAVE * 16 * WSTR]; /* Q tile (128*66) reused as P staging (8*16*84) */
  __shared__ float Ks[80 * KSTR];
  __shared__ float Vs[80 * VSTR];
  __shared__ int   smap[80];
  __shared__ float prm[NE_ * 4];

  const int lt = blockIdx.x, h = blockIdx.y, b = blockIdx.z;
  const int tid = threadIdx.x;
  const int lane = tid & 31, wave = tid >> 5;
  const int lm = lane & 15, half = lane >> 4, klo = half * 2;
  const int l0 = lt * LTILE;

  /* loads */
  const float* qbase = q + (((size_t)b * H_ + h) * L_ + l0) * D_;
  for (int i = 0; i < 8; ++i) {
    int fi = tid + i * 256;
    int row = fi >> 4, c4 = (fi & 15) * 4;
    float4 t = *(const float4*)(qbase + row * D_ + c4);
    float* d = &QW[row * QSTR + c4];
    d[0] = t.x; d[1] = t.y; d[2] = t.z; d[3] = t.w;
  }
  const float* kbase = kk_ + (((size_t)b * H_ + h) * S_) * D_;
  const float* vbase = vv_ + (((size_t)b * H_ + h) * S_) * D_;
  for (int i = 0; i < 5; ++i) {
    int fi = tid + i * 256;
    int row = fi >> 4, c4 = (fi & 15) * 4;
    float4 t = make_float4(0.f, 0.f, 0.f, 0.f);
    if (row < S_) t = *(const float4*)(kbase + row * D_ + c4);
    float* d = &Ks[row * KSTR + c4];
    d[0] = t.x; d[1] = t.y; d[2] = t.z; d[3] = t.w;
    t = make_float4(0.f, 0.f, 0.f, 0.f);
    if (row < S_) t = *(const float4*)(vbase + row * D_ + c4);
    d = &Vs[row * VSTR + c4];
    d[0] = t.x; d[1] = t.y; d[2] = t.z; d[3] = t.w;
  }
  if (tid == 0) {
    for (int s = 0; s < 80; ++s) smap[s] = -1;
    for (int j = 0; j < NE_; ++j)
      if (idx_b[b * NE_ + j] == b) {
        int s = idx_n[b * NE_ + j];
        if (s >= 0 && s < 80) smap[s] = j;
      }
  }
  if (tid < NE_ * 4) prm[tid] = params[(size_t)b * NE_ * 5 + (tid >> 2) * 5 + (tid & 3)];
  __syncthreads();

  const float invG = 1.f / gmax[0];

  /* scores: 16x80 per wave */
  v8f z = {0,0,0,0,0,0,0,0};
  v8f c[5]; c[0]=z; c[1]=z; c[2]=z; c[3]=z; c[4]=z;
  const float* qw = &QW[wave * 16 * QSTR];
  #pragma unroll
  for (int kstep = 0; kstep < D_; kstep += 4) {
    v2f a = { qw[lm * QSTR + kstep + klo], qw[lm * QSTR + kstep + klo + 1] };
    #pragma unroll
    for (int nb = 0; nb < 5; ++nb) {
      v2f bb = { Ks[(nb * 16 + lm) * KSTR + kstep + klo],
                 Ks[(nb * 16 + lm) * KSTR + kstep + klo + 1] };
      c[nb] = wmma_f32(a, bb, c[nb]);
    }
  }

  __syncthreads();  /* all waves done reading Q from QW before overwriting as P */

  /* softmax + gaussian scale, stage P into QW (A-layout source) */
  float* ww = &QW[wave * 16 * WSTR];
  #pragma unroll
  for (int vvv = 0; vvv < 8; ++vvv) {
    int rM = vvv + half * 8;
    int lrow = l0 + wave * 16 + rM;
    float sc[5];
    float m = -INFINITY;
    #pragma unroll
    for (int nb = 0; nb < 5; ++nb) {
      int n = nb * 16 + lm;
      sc[nb] = (n < S_) ? c[nb][vvv] * SCALE_ : -INFINITY;
      m = fmaxf(m, sc[nb]);
    }
    #pragma unroll
    for (int x = 1; x < 16; x <<= 1) m = fmaxf(m, __shfl_xor(m, x, 32));
    float e[5]; float sum = 0.f;
    #pragma unroll
    for (int nb = 0; nb < 5; ++nb) {
      int n = nb * 16 + lm;
      e[nb] = (n < S_) ? expf(sc[nb] - m) : 0.f;
      sum += e[nb];
    }
    #pragma unroll
    for (int x = 1; x < 16; x <<= 1) sum += __shfl_xor(sum, x, 32);
    float inv = 1.f / sum;
    float gx = (float)(lrow >> 6), gy = (float)(lrow & 63);  /* ref swaps axes here */
    #pragma unroll
    for (int nb = 0; nb < 5; ++nb) {
      int n = nb * 16 + lm;
      float w = e[nb] * inv;
      int j = smap[n];
      if (j >= 0) {
        float dx = gx - prm[j * 4 + 0];
        float dy = gy - prm[j * 4 + 1];
        w *= expf(-(dx * dx * prm[j * 4 + 2] + dy * dy * prm[j * 4 + 3])) * invG;
      }
      ww[rM * WSTR + n] = w;
    }
  }

  /* out = P(16x80) * V(80x64); wave-private LDS region, DS ops in-order per wave */
  v8f co[4]; co[0]=z; co[1]=z; co[2]=z; co[3]=z;
  #pragma unroll
  for (int kstep = 0; kstep < 80; kstep += 4) {
    v2f a = { ww[lm * WSTR + kstep + klo], ww[lm * WSTR + kstep + klo + 1] };
    #pragma unroll
    for (int nb = 0; nb < 4; ++nb) {
      v2f bb = { Vs[(kstep + klo) * VSTR + nb * 16 + lm],
                 Vs[(kstep + klo + 1) * VSTR + nb * 16 + lm] };
      co[nb] = wmma_f32(a, bb, co[nb]);
    }
  }
  float* ob = out + (((size_t)b * H_ + h) * L_ + l0 + wave * 16) * D_;
  #pragma unroll
  for (int vvv = 0; vvv < 8; ++vvv) {
    int rM = vvv + half * 8;
    #pragma unroll
    for (int nb = 0; nb < 4; ++nb) ob[rM * D_ + nb * 16 + lm] = co[nb][vvv];
  }
}

/* ---------------- launch ------------------------------------------------------ */
extern "C" void kernel_launch(void* const* d_in, const int* in_sizes, int n_in,
                              void* d_out, int out_size, void* d_ws, size_t ws_size,
                              hipStream_t stream) {
  const float* q   = (const float*)d_in[0];
  const float* k   = (const float*)d_in[1];
  const float* v   = (const float*)d_in[2];
  const int* idx_b = (const int*)d_in[3];
  const int* idx_n = (const int*)d_in[4];
  float* out = (float*)d_out;

  float* subj   = (float*)d_ws;                              /* B*L*NE  floats */
  float* params = subj + (size_t)B_ * L_ * NE_;              /* B*NE*5  floats */
  float* g      = params + (size_t)B_ * NE_ * 5;             /* 1 float        */

  dim3 g1(L_ / LTILE, B_);
  subj_scores_kernel<<<g1, 256, 0, stream>>>(q, k, idx_b, idx_n, subj);

  dim3 g2(NE_, B_);
  moments_kernel<<<g2, 256, 0, stream>>>(subj, params);

  gmax_kernel<<<1, 512, 0, stream>>>(params, g);

  dim3 g3(L_ / LTILE, H_, B_);
  attn_kernel<<<g3, 256, 0, stream>>>(q, k, v, idx_b, idx_n, params, g, out);
}